// Retention_69569880261100
// MI455X (gfx1250) — compile-verified
//
#include <hip/hip_runtime.h>
#include <hip/hip_bf16.h>
#include <math.h>

typedef __attribute__((ext_vector_type(2))) float v2f;
typedef __attribute__((ext_vector_type(8))) float v8f;

#define Dm    192
#define Hh    6
#define DKk   32
#define DVv   64
#define GSs   256
#define FFNf  384

static __device__ __forceinline__ v8f wmma4(v2f a, v2f b, v8f c) {
  // D = A(16x4,f32) * B(4x16,f32) + C(16x16,f32)
  return __builtin_amdgcn_wmma_f32_16x16x4_f32(false, a, false, b, (short)0, c,
                                               false, false);
}

// ---------------------------------------------------------------- sort keys
__global__ void build_keys(const int* __restrict__ coords,
                           unsigned long long* __restrict__ keys, int n) {
  int i = blockIdx.x * blockDim.x + threadIdx.x;
  if (i >= n) return;
  int b = coords[(size_t)i * 4 + 0];
  int x = coords[(size_t)i * 4 + 1];
  int y = coords[(size_t)i * 4 + 2];
  int z = coords[(size_t)i * 4 + 3];
  const int n1 = 41, m1 = 31;             // ceil(480/12)+1, ceil(360/12)+1
  int x1 = x / 12, y1 = y / 12, z1 = z / 4;
  int x2 = x % 12, y2 = y % 12, z2 = z % 4;
  int sy = ((y1 & 1) == 0) ? 1 : -1;
  int sx = ((x2 & 1) == 0) ? 1 : -1;
  int vx = (n1 * y1 + n1 * m1 * z1 + sy * x1) * (12 * 12 * 4) +
           sy * (12 * x2 + 12 * 12 * z2 + sx * y2);
  vx += b * (480 * 360 * 32 * 10);
  unsigned int uk = (unsigned int)vx ^ 0x80000000u;   // order-preserving for signed
  keys[i] = ((unsigned long long)uk << 32) | (unsigned int)i;  // stable tiebreak
}

__global__ void bitonic_step(unsigned long long* __restrict__ keys, int j, int k) {
  int i = blockIdx.x * blockDim.x + threadIdx.x;
  int ixj = i ^ j;
  if (ixj > i) {
    unsigned long long a = keys[i], b = keys[ixj];
    bool up = ((i & k) == 0);
    if ((a > b) == up) { keys[i] = b; keys[ixj] = a; }
  }
}

__global__ void extract_idx(const unsigned long long* __restrict__ keys,
                            int* __restrict__ idx, int n) {
  int i = blockIdx.x * blockDim.x + threadIdx.x;
  if (i < n) idx[i] = (int)(keys[i] & 0xffffffffull);
}

// ------------------------------------------------- gather + rms1 + pos-embed
__global__ void gather_prep(const float* __restrict__ inp,
                            const int* __restrict__ coords,
                            const int* __restrict__ idx,
                            const float* __restrict__ rn1w,
                            float* __restrict__ xs, float* __restrict__ hs,
                            float* __restrict__ qks, int4* __restrict__ csort) {
  int i = blockIdx.x;
  int t = threadIdx.x;  // 192 threads
  int j = idx[i];
  float x = inp[(size_t)j * Dm + t];
  float ss = x * x;
  for (int m = 16; m >= 1; m >>= 1) ss += __shfl_xor(ss, m, 32);
  __shared__ float part[6];
  __shared__ float stot;
  if ((t & 31) == 0) part[t >> 5] = ss;
  __syncthreads();
  if (t == 0) {
    float s = 0.f;
    for (int q = 0; q < 6; ++q) s += part[q];
    stot = s;
  }
  __syncthreads();
  float h = x * rsqrtf(stot / (float)Dm + 1e-6f) * rn1w[t];
  // pos embed: cols 0..63 from coords[:,0], 64..127 coords[:,1], 128..191 coords[:,2]
  int dimsel = t / 64;
  int r = t % 64;
  int tt = r >> 1;
  float vc = (float)coords[(size_t)j * 4 + dimsel];
  float freq = powf(10000.f, (float)tt / 32.f);
  float ang = vc / freq;
  float pe = (r & 1) ? cosf(ang) : sinf(ang);
  size_t o = (size_t)i * Dm + t;
  xs[o] = x;
  hs[o] = h;
  qks[o] = h + pe;
  if (t == 0) csort[i] = *(const int4*)(coords + (size_t)j * 4);
}

// ------------------------------------------------------ generic fp32 WMMA GEMM
// C[M x Nn] = alpha * A[M x K] * B[K x Nn] (+ R)    block: 192 thr, 64x96 tile
// LDS-staged: A chunk 64x32 (pad 34), B chunk transposed 96x32 (pad 34)
__global__ void gemm_wmma(const float* __restrict__ A, const float* __restrict__ B,
                          float* __restrict__ C, const float* __restrict__ R,
                          int M, int Nn, int K, float alpha) {
  __shared__ float sA[64 * 34];
  __shared__ float sBT[96 * 34];
  int tid = threadIdx.x;
  int w = tid >> 5;
  int lane = tid & 31;
  int half = lane >> 4;
  int l15 = lane & 15;
  int rw = w / 3, cw = w % 3;
  int row0b = blockIdx.y * 64;
  int col0b = blockIdx.x * 96;
  v8f acc[2][2];
  for (int r = 0; r < 2; ++r)
    for (int c = 0; c < 2; ++c) acc[r][c] = (v8f)0.f;

  for (int kc = 0; kc < K; kc += 32) {
    // stage A chunk: 64 rows x 32 k
    for (int idx2 = tid; idx2 < 64 * 32; idx2 += 192) {
      int row = idx2 >> 5, col = idx2 & 31;
      sA[row * 34 + col] = A[(size_t)(row0b + row) * K + kc + col];
    }
    // stage B chunk transposed: 32 k x 96 cols -> sBT[col][k]
    for (int idx2 = tid; idx2 < 32 * 96; idx2 += 192) {
      int kr = idx2 / 96, col = idx2 % 96;
      sBT[col * 34 + kr] = B[(size_t)(kc + kr) * Nn + col0b + col];
    }
    __syncthreads();
    for (int k0 = 0; k0 < 32; k0 += 4) {
      v2f a[2], b[2];
      for (int r = 0; r < 2; ++r)
        a[r] = *(const v2f*)(&sA[(rw * 32 + r * 16 + l15) * 34 + k0 + half * 2]);
      for (int c = 0; c < 2; ++c)
        b[c] = *(const v2f*)(&sBT[(cw * 32 + c * 16 + l15) * 34 + k0 + half * 2]);
      for (int r = 0; r < 2; ++r)
        for (int c = 0; c < 2; ++c) acc[r][c] = wmma4(a[r], b[c], acc[r][c]);
    }
    __syncthreads();
  }
  for (int r = 0; r < 2; ++r)
    for (int c = 0; c < 2; ++c)
      for (int i = 0; i < 8; ++i) {
        int row = row0b + rw * 32 + r * 16 + half * 8 + i;
        int col = col0b + cw * 32 + c * 16 + l15;
        float v = alpha * acc[r][c][i];
        if (R) v += R[(size_t)row * Nn + col];
        C[(size_t)row * Nn + col] = v;
      }
}

// -------------------------------------------- retention attention, per (g,h)
// Stages K (256x32) and V^T (64x256) once in LDS, loops over 8 row slabs.
__global__ void attn_kernel(const float* __restrict__ q, const float* __restrict__ kk,
                            const float* __restrict__ v, const int4* __restrict__ csort,
                            float* __restrict__ o) {
  int g = blockIdx.x;
  int hh = blockIdx.y;
  int tid = threadIdx.x;       // 256
  __shared__ float sK[256 * 34];    // k rows, pad-34
  __shared__ float sVT[64 * 258];   // V transposed: [col][k], pad-258
  __shared__ float sS[32 * 256];
  __shared__ int scx[256], scy[256], scz[256];
  {
    int4 c = csort[(size_t)g * GSs + tid];
    scx[tid] = c.y;
    scy[tid] = c.z;
    scz[tid] = c.w;
    for (int idx2 = tid; idx2 < 256 * 32; idx2 += 256) {
      int row = idx2 >> 5, col = idx2 & 31;
      sK[row * 34 + col] = kk[(size_t)(g * GSs + row) * Dm + hh * DKk + col];
    }
    for (int idx2 = tid; idx2 < 64 * 256; idx2 += 256) {
      int krow = idx2 >> 6, col = idx2 & 63;
      sVT[col * 258 + krow] = v[(size_t)(g * GSs + krow) * (2 * Dm) + hh * DVv + col];
    }
  }
  __syncthreads();
  int w = tid >> 5, lane = tid & 31, half = lane >> 4, l15 = lane & 15;
  int r = w >> 2;
  float decay = logf(1.f - exp2f(-5.f - (float)hh));

  for (int rb = 0; rb < 8; ++rb) {
    // ---- phase 1: S = q k^T (k pre-scaled by DK^-0.5)
    {
      int rowg = g * GSs + rb * 32 + r * 16;
      for (int jt = 0; jt < 4; ++jt) {
        int ct = (w & 3) * 4 + jt;
        v8f acc = (v8f)0.f;
        for (int k0 = 0; k0 < DKk; k0 += 4) {
          v2f a = *(const v2f*)(q + (size_t)(rowg + l15) * Dm + hh * DKk + k0 + half * 2);
          v2f b = *(const v2f*)(&sK[(ct * 16 + l15) * 34 + k0 + half * 2]);
          acc = wmma4(a, b, acc);
        }
        for (int i = 0; i < 8; ++i)
          sS[(r * 16 + half * 8 + i) * 256 + ct * 16 + l15] = acc[i];
      }
    }
    __syncthreads();
    // ---- phase 2: softmax * decay mask (8 lanes per row)
    {
      int sl = tid >> 3, sub = tid & 7;
      int rloc = rb * 32 + sl;
      int cx = scx[rloc], cy = scy[rloc], cz = scz[rloc];
      float m = -3.4e38f;
      for (int t = sub; t < 256; t += 8) m = fmaxf(m, sS[sl * 256 + t]);
      for (int mm = 4; mm >= 1; mm >>= 1) m = fmaxf(m, __shfl_xor(m, mm, 8));
      float sumP = 0.f, dsum = 0.f;
      for (int t = sub; t < 256; t += 8) {
        float s = sS[sl * 256 + t];
        float p = expf(s - m);
        int dist = abs(cx - scx[t]) + abs(cy - scy[t]) + abs(cz - scz[t]);
        float dmv = expf(decay * (float)dist);
        sumP += p;
        dsum += dmv;
        sS[sl * 256 + t] = p * dmv;
      }
      for (int mm = 4; mm >= 1; mm >>= 1) {
        sumP += __shfl_xor(sumP, mm, 8);
        dsum += __shfl_xor(dsum, mm, 8);
      }
      float sc = 1.f / (sumP * sqrtf(dsum));
      for (int t = sub; t < 256; t += 8) sS[sl * 256 + t] *= sc;
    }
    __syncthreads();
    // ---- phase 3: O = A * V   (2 row tiles x 4 col tiles over DV=64)
    {
      int c = w & 3;
      v8f acc = (v8f)0.f;
      for (int k0 = 0; k0 < 256; k0 += 4) {
        v2f a = *(const v2f*)(&sS[(r * 16 + l15) * 256 + k0 + half * 2]);
        v2f b = *(const v2f*)(&sVT[(c * 16 + l15) * 258 + k0 + half * 2]);
        acc = wmma4(a, b, acc);
      }
      for (int i = 0; i < 8; ++i)
        o[(size_t)(g * GSs + rb * 32 + r * 16 + half * 8 + i) * (2 * Dm) + hh * DVv + c * 16 + l15] = acc[i];
    }
    __syncthreads();
  }
}

// ------------------------------------------- per-head RMS * gn_w * silu(gate)
__global__ void rms_head_silu(float* __restrict__ o, const float* __restrict__ g,
                              const float* __restrict__ gnw) {
  int row = blockIdx.x, hh = blockIdx.y, ci = threadIdx.x;  // 64 threads
  size_t off = (size_t)row * (2 * Dm) + hh * DVv + ci;
  float val = o[off];
  float ss = val * val;
  for (int m = 16; m >= 1; m >>= 1) ss += __shfl_xor(ss, m, 32);
  __shared__ float part[2];
  if ((ci & 31) == 0) part[ci >> 5] = ss;
  __syncthreads();
  float tot = part[0] + part[1];
  float sc = rsqrtf(tot / (float)DVv + 1e-6f);
  float gv = g[off];
  float sil = gv / (1.f + expf(-gv));
  o[off] = val * sc * gnw[ci] * sil;
}

// ------------------------------------------------------------- rms over D=192
__global__ void rms_row(const float* __restrict__ x, const float* __restrict__ w,
                        float* __restrict__ out) {
  int i = blockIdx.x, t = threadIdx.x;  // 192 threads
  float v = x[(size_t)i * Dm + t];
  float ss = v * v;
  for (int m = 16; m >= 1; m >>= 1) ss += __shfl_xor(ss, m, 32);
  __shared__ float part[6];
  __shared__ float stot;
  if ((t & 31) == 0) part[t >> 5] = ss;
  __syncthreads();
  if (t == 0) {
    float s = 0.f;
    for (int q2 = 0; q2 < 6; ++q2) s += part[q2];
    stot = s;
  }
  __syncthreads();
  out[(size_t)i * Dm + t] = v * rsqrtf(stot / (float)Dm + 1e-6f) * w[t];
}

// ------------------------------------------------- f = gelu_exact(a) * gate
__global__ void gelu_mul(float* __restrict__ a, const float* __restrict__ g, size_t n) {
  size_t t = (size_t)blockIdx.x * blockDim.x + threadIdx.x;
  if (t >= n) return;
  float x = a[t];
  float ge = 0.5f * x * (1.f + erff(x * 0.70710678118654752f));
  a[t] = ge * g[t];
}

// --------------------------------------------------------- scatter via idx
__global__ void scatter_rows(const float* __restrict__ y, const int* __restrict__ idx,
                             float* __restrict__ out, size_t n) {
  size_t t = (size_t)blockIdx.x * blockDim.x + threadIdx.x;
  if (t >= n) return;
  size_t i = t / Dm, c = t % Dm;
  out[(size_t)idx[i] * Dm + c] = y[t];
}

extern "C" void kernel_launch(void* const* d_in, const int* in_sizes, int n_in,
                              void* d_out, int out_size, void* d_ws, size_t ws_size,
                              hipStream_t stream) {
  const float* inp    = (const float*)d_in[0];
  const int*   coords = (const int*)d_in[1];
  const float* Wq     = (const float*)d_in[3];
  const float* Wk     = (const float*)d_in[4];
  const float* Wv     = (const float*)d_in[5];
  const float* Wg     = (const float*)d_in[6];
  const float* Wout   = (const float*)d_in[7];
  const float* gn_w   = (const float*)d_in[8];
  const float* rn1_w  = (const float*)d_in[9];
  const float* rn2_w  = (const float*)d_in[10];
  const float* W1     = (const float*)d_in[11];
  const float* Wgate  = (const float*)d_in[12];
  const float* W2     = (const float*)d_in[13];
  float* out = (float*)d_out;

  const size_t N = (size_t)in_sizes[0] / Dm;   // 65536
  const int Ni = (int)N;
  const size_t f192 = N * Dm * sizeof(float);
  const size_t f384 = N * 2 * Dm * sizeof(float);

  char* p = (char*)d_ws;
  float* B0 = (float*)p; p += f192;        // x_sorted (resid1)
  float* B1 = (float*)p; p += f192;        // h   -> x1
  float* B2 = (float*)p; p += f192;        // qk  -> h2
  float* B3 = (float*)p; p += f192;        // q   -> y
  float* B4 = (float*)p; p += f192;        // k
  float* B5 = (float*)p; p += f384;        // v   -> a1 -> f
  float* B6 = (float*)p; p += f384;        // g   -> g1
  float* B7 = (float*)p; p += f384;        // o
  unsigned long long* keys = (unsigned long long*)p; p += N * 8;
  int*  idx   = (int*)p;  p += N * 4;
  int4* csort = (int4*)p; p += N * 16;
  (void)ws_size; (void)n_in; (void)out_size;

  // 1) sort keys + stable bitonic argsort
  build_keys<<<(Ni + 255) / 256, 256, 0, stream>>>(coords, keys, Ni);
  for (unsigned k = 2; k <= N; k <<= 1)
    for (unsigned j = k >> 1; j > 0; j >>= 1)
      bitonic_step<<<Ni / 256, 256, 0, stream>>>(keys, (int)j, (int)k);
  extract_idx<<<(Ni + 255) / 256, 256, 0, stream>>>(keys, idx, Ni);

  // 2) gather + rms1 + pos embed
  gather_prep<<<Ni, Dm, 0, stream>>>(inp, coords, idx, rn1_w, B0, B1, B2, csort);

  auto gemm = [&](const float* A, const float* Bm, float* C, const float* R,
                  int M, int Nn, int K, float alpha) {
    dim3 gr(Nn / 96, M / 64);
    gemm_wmma<<<gr, 192, 0, stream>>>(A, Bm, C, R, M, Nn, K, alpha);
  };

  // 3) projections (fp32 WMMA)
  gemm(B2, Wq, B3, nullptr, Ni, Dm, Dm, 1.0f);                    // q
  gemm(B2, Wk, B4, nullptr, Ni, Dm, Dm, 0.17677669529663689f);    // k * DK^-0.5
  gemm(B1, Wv, B5, nullptr, Ni, 2 * Dm, Dm, 1.0f);                // v
  gemm(B1, Wg, B6, nullptr, Ni, 2 * Dm, Dm, 1.0f);                // gate

  // 4) retention attention: one block per (group, head)
  {
    dim3 gr((unsigned)(N / GSs), Hh);
    attn_kernel<<<gr, 256, 0, stream>>>(B3, B4, B5, csort, B7);
  }

  // 5) per-head rms * gn_w * silu(gate), in place on o
  {
    dim3 gr((unsigned)N, Hh);
    rms_head_silu<<<gr, DVv, 0, stream>>>(B7, B6, gn_w);
  }

  // 6) x1 = o2 @ Wout + resid
  gemm(B7, Wout, B1, B0, Ni, Dm, 2 * Dm, 1.0f);

  // 7) FFN
  rms_row<<<Ni, Dm, 0, stream>>>(B1, rn2_w, B2);                  // h2
  gemm(B2, W1, B5, nullptr, Ni, 2 * Dm, Dm, 1.0f);                // a1
  gemm(B2, Wgate, B6, nullptr, Ni, 2 * Dm, Dm, 1.0f);             // g1
  {
    size_t n = N * 2 * Dm;
    gelu_mul<<<(unsigned)((n + 255) / 256), 256, 0, stream>>>(B5, B6, n);
  }
  gemm(B5, W2, B3, B1, Ni, Dm, 2 * Dm, 1.0f);                     // y

  // 8) scatter back through permutation
  {
    size_t n = N * Dm;
    scatter_rows<<<(unsigned)((n + 255) / 256), 256, 0, stream>>>(B3, idx, out, n);
  }
}